// quadratic_spline_57320633532673
// MI455X (gfx1250) — compile-verified
//
#include <hip/hip_runtime.h>
#include <stdint.h>

#define N_CHANNELS 64
#define N_KNOTS    101
#define HALF       50            // N_KNOTS / 2
#define COEF_LEN   (N_CHANNELS * N_KNOTS)   // 6464 floats = 25856 B
#define BLOCK      256

typedef float v4f __attribute__((ext_vector_type(4)));   // true clang vector type

// x layout: [B=16, C=64, H=256, W=256]; HW = 65536 elements per (b,c) slice.
// Processing 4-float vectors: 16384 vectors per slice -> channel of vector v is
// (v >> 14) & 63, constant across a wave.

__global__ __launch_bounds__(BLOCK) void quadratic_spline_kernel(
    const float* __restrict__ x,
    const float* __restrict__ coef,
    const float* __restrict__ grid_p,
    const int*   __restrict__ zk,
    float*       __restrict__ out,
    int nvec)
{
    __shared__ float s_coef[COEF_LEN];
    __shared__ int   s_zk[N_CHANNELS];

    // ---- Stage coefficient table into LDS via CDNA5 async global->LDS DMA ----
    // global_load_async_to_lds_b32: VDST = LDS byte address (low 32 bits of the
    // flat LDS pointer), VADDR = 64-bit global address, no SADDR ("off").
    // Tracked by ASYNCcnt.
    for (int i = threadIdx.x; i < COEF_LEN; i += BLOCK) {
        unsigned lds_addr = (unsigned)(uintptr_t)(&s_coef[i]);
        const float* gp = coef + i;
        asm volatile("global_load_async_to_lds_b32 %0, %1, off"
                     :: "v"(lds_addr), "v"(gp) : "memory");
    }
    if (threadIdx.x < N_CHANNELS) {
        unsigned lds_addr = (unsigned)(uintptr_t)(&s_zk[threadIdx.x]);
        const int* gp = zk + (int)threadIdx.x;
        asm volatile("global_load_async_to_lds_b32 %0, %1, off"
                     :: "v"(lds_addr), "v"(gp) : "memory");
    }
    asm volatile("s_wait_asynccnt 0" ::: "memory");
    __syncthreads();

    const float g     = *grid_p;          // scalar, L2/constant-cached
    const float inv_g = 1.0f / g;         // divide once; multiply per element
    const float lo    = -g * (float)HALF;
    const float hi    =  g * (float)(HALF - 2);

    const int tid    = blockIdx.x * BLOCK + threadIdx.x;
    const int stride = gridDim.x * BLOCK;

    for (int v = tid; v < nvec; v += stride) {
        // Streamed once: non-temporal b128 load.
        const v4f xv = __builtin_nontemporal_load(((const v4f*)x) + v);
        const int c    = (v >> 14) & (N_CHANNELS - 1);
        const int base = s_zk[c];         // wave-uniform

        v4f ov;
        #pragma unroll
        for (int k = 0; k < 4; ++k) {
            const float xi = xv[k];
            // clamp, bucket index
            float xc = fminf(fmaxf(xi, lo), hi);
            float f  = floorf(xc * inv_g);
            int  idx = base + (int)f;
            // fractional position (uses UNclamped xi, per reference)
            float sh = (xi - f * g) * inv_g;
            float s2 = sh * sh;
            float frac3 = 0.5f * s2;                              // sh^2/2
            float frac2 = 0.5f * (-2.0f * s2 + 2.0f * sh + 1.0f); // (-2sh^2+2sh+1)/2
            float frac1 = 0.5f * (s2 - 2.0f * sh + 1.0f);         // (sh-1)^2/2
            // three LDS gathers + combine
            ov[k] = s_coef[idx + 2] * frac3
                  + s_coef[idx + 1] * frac2
                  + s_coef[idx]     * frac1;
        }
        __builtin_nontemporal_store(ov, ((v4f*)out) + v);
    }
}

extern "C" void kernel_launch(void* const* d_in, const int* in_sizes, int n_in,
                              void* d_out, int out_size, void* d_ws, size_t ws_size,
                              hipStream_t stream) {
    const float* x    = (const float*)d_in[0];
    const float* coef = (const float*)d_in[1];
    const float* grid = (const float*)d_in[2];
    const int*   zk   = (const int*)d_in[3];
    float*       out  = (float*)d_out;

    const int n    = in_sizes[0];      // 16*64*256*256, divisible by 4
    const int nvec = n / 4;

    int blocks = (nvec + BLOCK - 1) / BLOCK;
    if (blocks > 4096) blocks = 4096;  // grid-stride; ~32K wave32 waves in flight

    quadratic_spline_kernel<<<blocks, BLOCK, 0, stream>>>(x, coef, grid, zk, out, nvec);
}